// Qwen3OmniMoeThinkerTextSparseMoeBlock_37520834298112
// MI455X (gfx1250) — compile-verified
//
#include <hip/hip_runtime.h>
#include <math.h>

constexpr int NT = 512;    // tokens
constexpr int ND = 2048;   // hidden
constexpr int NE = 64;     // experts
constexpr int NK = 8;      // top-k
constexpr int NF = 768;    // moe intermediate

typedef float v2f __attribute__((ext_vector_type(2)));
typedef float v8f __attribute__((ext_vector_type(8)));

static __device__ __forceinline__ v8f wmma_f32_k4(v2f a, v2f b, v8f c) {
  // D(16x16,f32) = A(16x4,f32) * B(4x16,f32) + C
  return __builtin_amdgcn_wmma_f32_16x16x4_f32(false, a, false, b, (short)0, c, false, false);
}

// ---------------- router: logits -> top-8 -> renormalized weights ----------------
__global__ void init_cnt_kernel(int* __restrict__ cnt) {
  int i = threadIdx.x;
  if (i < NE) cnt[i] = 0;
}

__global__ void router_kernel(const float* __restrict__ x,
                              const float* __restrict__ gw,      // [D, E]
                              int* __restrict__ topk_idx,        // [T, K]
                              float* __restrict__ topk_w,        // [T, K]
                              int* __restrict__ cnt) {           // [E]
  __shared__ float sl[NE];
  const int t = blockIdx.x;
  const int e = threadIdx.x;
  const float* xr = x + (size_t)t * ND;
  float acc = 0.f;
  for (int d = 0; d < ND; ++d) acc = fmaf(xr[d], gw[(size_t)d * NE + e], acc);
  sl[e] = acc;
  __syncthreads();
  if (e == 0) {
    // softmax -> top-k -> renorm  ==  exp(top-k logits - max) / sum over the k
    float mx = -INFINITY;
    for (int i = 0; i < NE; ++i) mx = fmaxf(mx, sl[i]);
    int idxs[NK]; float vals[NK];
    float s = 0.f;
    for (int k = 0; k < NK; ++k) {
      int bi = 0; float bv = -INFINITY;
      for (int i = 0; i < NE; ++i) { float v = sl[i]; if (v > bv) { bv = v; bi = i; } }
      sl[bi] = -INFINITY;                 // consume
      idxs[k] = bi;
      float ev = __expf(bv - mx);
      vals[k] = ev; s += ev;
    }
    float inv = 1.f / s;
    for (int k = 0; k < NK; ++k) {
      topk_idx[t * NK + k] = idxs[k];
      topk_w[t * NK + k]   = vals[k] * inv;
      atomicAdd(&cnt[idxs[k]], 1);
    }
  }
}

// ---------------- prefix scan over expert counts ----------------
__global__ void scan_kernel(const int* __restrict__ cnt,
                            int* __restrict__ offs,   // [E+1]
                            int* __restrict__ cur) {  // [E]
  if (threadIdx.x == 0 && blockIdx.x == 0) {
    int a = 0;
    for (int e = 0; e < NE; ++e) { offs[e] = a; a += cnt[e]; cur[e] = 0; }
    offs[NE] = a;   // == T*K
  }
}

// ---------------- assign (token,k) pairs to expert row lists ----------------
__global__ void assign_kernel(const int* __restrict__ topk_idx,
                              const int* __restrict__ offs,
                              int* __restrict__ cur,
                              int* __restrict__ rows) {   // [T*K] pair-ids grouped by expert
  int pid = blockIdx.x * blockDim.x + threadIdx.x;
  if (pid >= NT * NK) return;
  int e = topk_idx[pid];
  int pos = atomicAdd(&cur[e], 1);
  rows[offs[e] + pos] = pid;               // pair id = t*NK + k
}

// ---------------- gate+up GEMMs fused (share A), SiLU*up -> hbuf ----------------
// one wave per (expert, 16-wide F strip, 64-row M-block); B fragments reused 4x
__global__ void gateup_kernel(const float* __restrict__ x,      // [T, D]
                              const float* __restrict__ wg,     // [E, D, F]
                              const float* __restrict__ wu,     // [E, D, F]
                              const int* __restrict__ rows,
                              const int* __restrict__ offs,
                              float* __restrict__ hbuf) {       // [T*K, F]
  const int ntiles = NF / 16;
  const int e  = blockIdx.x / ntiles;
  const int n0 = (blockIdx.x % ntiles) * 16;
  const int base  = offs[e];
  const int cnt_e = offs[e + 1] - base;
  const int m0 = blockIdx.y * 64;
  if (m0 >= cnt_e) return;
  const int lane = threadIdx.x;
  const int half = lane >> 4;     // 0: K0/K1, 1: K2/K3 (A & B fragment halves)
  const int l16  = lane & 15;
  const float* wg_e = wg + (size_t)e * ND * NF;
  const float* wu_e = wu + (size_t)e * ND * NF;

  int tok[4];
#pragma unroll
  for (int s = 0; s < 4; ++s) {
    int r = m0 + s * 16 + l16;
    if (r >= cnt_e) r = cnt_e - 1;          // clamp padding rows (results discarded)
    tok[s] = rows[base + r] / NK;           // token id
  }
  v8f accG[4], accU[4];
#pragma unroll
  for (int s = 0; s < 4; ++s)
#pragma unroll
    for (int j = 0; j < 8; ++j) { accG[s][j] = 0.f; accU[s][j] = 0.f; }

  for (int k0 = 0; k0 < ND; k0 += 4) {
    const int ka = k0 + 2 * half;
    v2f bg, bu;
    bg.x = wg_e[(size_t)ka * NF + n0 + l16];
    bg.y = wg_e[(size_t)(ka + 1) * NF + n0 + l16];
    bu.x = wu_e[(size_t)ka * NF + n0 + l16];
    bu.y = wu_e[(size_t)(ka + 1) * NF + n0 + l16];
#pragma unroll
    for (int s = 0; s < 4; ++s) {
      const float* ap = x + (size_t)tok[s] * ND + ka;
      v2f a; a.x = ap[0]; a.y = ap[1];
      accG[s] = wmma_f32_k4(a, bg, accG[s]);
      accU[s] = wmma_f32_k4(a, bu, accU[s]);
    }
  }
  // h = silu(g) * u ; C/D layout: VGPR j -> row j (lanes 0-15) / j+8 (lanes 16-31)
#pragma unroll
  for (int s = 0; s < 4; ++s) {
#pragma unroll
    for (int j = 0; j < 8; ++j) {
      int r = m0 + s * 16 + half * 8 + j;
      if (r < cnt_e) {
        float g = accG[s][j], u = accU[s][j];
        float hv = g * __builtin_amdgcn_rcpf(1.f + __expf(-g)) * u;  // v_rcp_f32, no IEEE div
        hbuf[(size_t)(base + r) * NF + n0 + l16] = hv;
      }
    }
  }
}

// ---------------- down GEMM, scale by routing weight, scatter to contrib ----------------
__global__ void down_kernel(const float* __restrict__ hbuf,    // [T*K, F]
                            const float* __restrict__ wd,      // [E, F, D]
                            const int* __restrict__ rows,
                            const int* __restrict__ offs,
                            const float* __restrict__ topk_w,  // [T*K]
                            float* __restrict__ contrib) {     // [T*K, D]
  const int ntiles = ND / 16;
  const int e  = blockIdx.x / ntiles;
  const int n0 = (blockIdx.x % ntiles) * 16;
  const int base  = offs[e];
  const int cnt_e = offs[e + 1] - base;
  const int m0 = blockIdx.y * 64;
  if (m0 >= cnt_e) return;
  const int lane = threadIdx.x;
  const int half = lane >> 4;
  const int l16  = lane & 15;
  const float* wd_e = wd + (size_t)e * NF * ND;

  int arow[4];
#pragma unroll
  for (int s = 0; s < 4; ++s) {
    int r = m0 + s * 16 + l16;
    if (r >= cnt_e) r = cnt_e - 1;
    arow[s] = base + r;
  }
  v8f acc[4];
#pragma unroll
  for (int s = 0; s < 4; ++s)
#pragma unroll
    for (int j = 0; j < 8; ++j) acc[s][j] = 0.f;

  for (int k0 = 0; k0 < NF; k0 += 4) {
    const int ka = k0 + 2 * half;
    v2f b;
    b.x = wd_e[(size_t)ka * ND + n0 + l16];
    b.y = wd_e[(size_t)(ka + 1) * ND + n0 + l16];
#pragma unroll
    for (int s = 0; s < 4; ++s) {
      const float* ap = hbuf + (size_t)arow[s] * NF + ka;
      v2f a; a.x = ap[0]; a.y = ap[1];
      acc[s] = wmma_f32_k4(a, b, acc[s]);
    }
  }
#pragma unroll
  for (int s = 0; s < 4; ++s) {
#pragma unroll
    for (int j = 0; j < 8; ++j) {
      int r = m0 + s * 16 + half * 8 + j;
      if (r < cnt_e) {
        int pair = rows[base + r];
        float w = topk_w[pair];
        contrib[(size_t)pair * ND + n0 + l16] = w * acc[s][j];
      }
    }
  }
}

// ---------------- deterministic combine over the 8 slots per token (float4) ----------------
__global__ void combine_kernel(const float* __restrict__ contrib,
                               float* __restrict__ out) {       // [T, D]
  int idx = blockIdx.x * blockDim.x + threadIdx.x;      // over T*D/4
  if (idx >= NT * ND / 4) return;
  int t  = idx / (ND / 4);
  int d4 = idx - t * (ND / 4);
  float4 s = make_float4(0.f, 0.f, 0.f, 0.f);
#pragma unroll
  for (int k = 0; k < NK; ++k) {
    const float4* src = (const float4*)(contrib + (size_t)(t * NK + k) * ND) + d4;
    float4 v = *src;
    s.x += v.x; s.y += v.y; s.z += v.z; s.w += v.w;
  }
  ((float4*)(out + (size_t)t * ND))[d4] = s;
}

extern "C" void kernel_launch(void* const* d_in, const int* in_sizes, int n_in,
                              void* d_out, int out_size, void* d_ws, size_t ws_size,
                              hipStream_t stream) {
  (void)in_sizes; (void)n_in; (void)out_size; (void)ws_size;
  const float* x  = (const float*)d_in[0];   // [T, D]
  const float* gw = (const float*)d_in[1];   // [D, E]
  const float* wg = (const float*)d_in[2];   // [E, D, F]
  const float* wu = (const float*)d_in[3];   // [E, D, F]
  const float* wd = (const float*)d_in[4];   // [E, F, D]
  float* out = (float*)d_out;

  // workspace carve-up (256B aligned sections), total ~45 MB
  char* p = (char*)d_ws;
  auto carve = [&](size_t bytes) {
    char* r = p;
    p += (bytes + 255) & ~(size_t)255;
    return r;
  };
  int*   topk_idx = (int*)  carve((size_t)NT * NK * sizeof(int));
  float* topk_w   = (float*)carve((size_t)NT * NK * sizeof(float));
  int*   cnt      = (int*)  carve((size_t)NE * sizeof(int));
  int*   offs     = (int*)  carve((size_t)(NE + 1) * sizeof(int));
  int*   cur      = (int*)  carve((size_t)NE * sizeof(int));
  int*   rows     = (int*)  carve((size_t)NT * NK * sizeof(int));
  float* hbuf     = (float*)carve((size_t)NT * NK * NF * sizeof(float));   // 12.6 MB
  float* contrib  = (float*)carve((size_t)NT * NK * ND * sizeof(float));   // 32 MB

  init_cnt_kernel<<<1, 64, 0, stream>>>(cnt);
  router_kernel<<<NT, NE, 0, stream>>>(x, gw, topk_idx, topk_w, cnt);
  scan_kernel<<<1, 1, 0, stream>>>(cnt, offs, cur);
  assign_kernel<<<(NT * NK + 255) / 256, 256, 0, stream>>>(topk_idx, offs, cur, rows);
  // grid.y = max M-blocks (512/64 = 8); blocks past cnt_e exit immediately
  gateup_kernel<<<dim3(NE * (NF / 16), 8), 32, 0, stream>>>(x, wg, wu, rows, offs, hbuf);
  down_kernel<<<dim3(NE * (ND / 16), 8), 32, 0, stream>>>(hbuf, wd, rows, offs, topk_w, contrib);
  combine_kernel<<<(NT * ND / 4 + 255) / 256, 256, 0, stream>>>(contrib, out);
}